// HashGrid_50328426775011
// MI455X (gfx1250) — compile-verified
//
#include <hip/hip_runtime.h>
#include <cstdint>

// Instant-NGP hash-grid encode, fused over all 10 LODs.
// All codebooks (5255 entries, padded to float4 = 84,080 B) live in LDS,
// staged via gfx1250 async global->LDS loads (ASYNCcnt).

#define NLOD  10
#define BLOCK 256

namespace {
constexpr int      kRes [NLOD] = {7, 8, 11, 14, 18, 23, 30, 38, 50, 65};
constexpr unsigned kSize[NLOD] = {49u, 64u, 121u, 196u, 324u, 529u, 900u,
                                  1024u, 1024u, 1024u};
constexpr int      kBase[NLOD] = {0, 49, 113, 234, 430, 754, 1283,
                                  2183, 3207, 4231};
constexpr int      kTotalEntries = 5255;      // sum of kSize
constexpr unsigned P1 = 2654435761u;          // hash primes (PRIMES[1], [2])
constexpr unsigned P2 = 805459861u;
}

struct CBPtrs { const float* p[NLOD]; };

typedef __attribute__((address_space(1))) int gint_t;   // global
typedef __attribute__((address_space(3))) int lint_t;   // LDS

// One 4-byte async DMA global -> LDS (falls back to load+ds_store if the
// builtin is absent on this toolchain).
__device__ __forceinline__ void g2l_async_b32(const float* g, float* l) {
#if defined(__gfx1250__) && __has_builtin(__builtin_amdgcn_global_load_async_to_lds_b32)
  __builtin_amdgcn_global_load_async_to_lds_b32(
      (gint_t*)(uintptr_t)g,
      (lint_t*)(uint32_t)(uintptr_t)l,   // generic LDS addr low 32b = LDS offset
      0, 0);
#else
  *l = *g;
#endif
}

__device__ __forceinline__ void wait_async_all() {
#if defined(__gfx1250__) && __has_builtin(__builtin_amdgcn_s_wait_asynccnt)
  __builtin_amdgcn_s_wait_asynccnt(0);
#elif defined(__gfx1250__)
  asm volatile("s_wait_asynccnt 0" ::: "memory");
#endif
}

__global__ __launch_bounds__(BLOCK)
void hashgrid_fused_kernel(const float* __restrict__ pts, CBPtrs cbs,
                           float* __restrict__ out, int n)
{
  // Entries padded to float4 (w unused) so each corner is one ds_load_b128.
  __shared__ float4 lut[kTotalEntries];     // 84,080 bytes < 320KB/WGP

  // ---- stage all codebooks into LDS (async DMA path) ----
  for (int l = 0; l < NLOD; ++l) {
    const float* __restrict__ src = cbs.p[l];
    float* dst0 = reinterpret_cast<float*>(&lut[kBase[l]]);
    const int sz = (int)kSize[l];
    for (int e = (int)threadIdx.x; e < sz; e += BLOCK) {
      const float* s = src + 3 * e;
      float*       d = dst0 + 4 * e;
      g2l_async_b32(s + 0, d + 0);
      g2l_async_b32(s + 1, d + 1);
      g2l_async_b32(s + 2, d + 2);
    }
  }
  wait_async_all();     // drain this wave's ASYNCcnt
  __syncthreads();      // make LDS visible across all 8 waves

  const int stride = (int)(gridDim.x * blockDim.x);
  for (int i = (int)(blockIdx.x * blockDim.x + threadIdx.x); i < n; i += stride) {
    // next-iteration point prefetch (global_prefetch_b8)
    if (i + stride < n)
      __builtin_prefetch(pts + 3 * (i + stride), 0, 0);

    // streamed exactly once -> non-temporal
    const float px = __builtin_nontemporal_load(pts + 3 * i + 0);
    const float py = __builtin_nontemporal_load(pts + 3 * i + 1);
    const float pz = __builtin_nontemporal_load(pts + 3 * i + 2);

    float a0 = 0.f, a1 = 0.f, a2 = 0.f;

#pragma unroll
    for (int l = 0; l < NLOD; ++l) {
      const float rm1 = (float)(kRes[l] - 1);
      const float xs = px * rm1, ys = py * rm1, zs = pz * rm1;

      int x0 = (int)xs; if (x0 > kRes[l] - 2) x0 = kRes[l] - 2;
      int y0 = (int)ys; if (y0 > kRes[l] - 2) y0 = kRes[l] - 2;
      int z0 = (int)zs; if (z0 > kRes[l] - 2) z0 = kRes[l] - 2;

      const float fx = xs - (float)x0;
      const float fy = ys - (float)y0;
      const float fz = zs - (float)z0;
      const float wx[2] = {1.f - fx, fx};
      const float wy[2] = {1.f - fy, fy};
      const float wz[2] = {1.f - fz, fz};

      // per-axis hash terms; corner +1 is just "+prime" pre-XOR
      const unsigned hx[2] = {(unsigned)x0, (unsigned)x0 + 1u};
      const unsigned hyb = (unsigned)y0 * P1;
      const unsigned hy[2] = {hyb, hyb + P1};
      const unsigned hzb = (unsigned)z0 * P2;
      const unsigned hz[2] = {hzb, hzb + P2};

#pragma unroll
      for (int c = 0; c < 8; ++c) {
        const int bx = c & 1, by = (c >> 1) & 1, bz = (c >> 2) & 1;
        const unsigned h = hx[bx] ^ hy[by] ^ hz[bz];
        const int idx = (int)(h % kSize[l]);        // constant modulo -> mulhi
        const float4 f = lut[kBase[l] + idx];       // ds_load_b128
        const float  w = wx[bx] * wy[by] * wz[bz];
        a0 = fmaf(w, f.x, a0);
        a1 = fmaf(w, f.y, a1);
        a2 = fmaf(w, f.z, a2);
      }
    }

    __builtin_nontemporal_store(a0, out + 3 * i + 0);
    __builtin_nontemporal_store(a1, out + 3 * i + 1);
    __builtin_nontemporal_store(a2, out + 3 * i + 2);
  }
}

extern "C" void kernel_launch(void* const* d_in, const int* in_sizes, int n_in,
                              void* d_out, int out_size, void* d_ws, size_t ws_size,
                              hipStream_t stream) {
  (void)d_ws; (void)ws_size; (void)out_size; (void)n_in;
  const float* pts = (const float*)d_in[0];
  CBPtrs cbs;
  for (int l = 0; l < NLOD; ++l)
    cbs.p[l] = (const float*)d_in[1 + l];
  const int n = in_sizes[0] / 3;            // 2,000,000 points

  const int ptsPerThread = 8;               // amortize LDS fill
  const int ptsPerBlock  = BLOCK * ptsPerThread;
  int grid = (n + ptsPerBlock - 1) / ptsPerBlock;
  if (grid < 1) grid = 1;

  hipLaunchKernelGGL(hashgrid_fused_kernel, dim3(grid), dim3(BLOCK), 0, stream,
                     pts, cbs, (float*)d_out, n);
}